// GCN_74122545594672
// MI455X (gfx1250) — compile-verified
//
#include <hip/hip_runtime.h>
#include <hip/hip_bf16.h>

#define Hh 64          // hidden size
#define WAVE 32

typedef __attribute__((ext_vector_type(2))) float v2f;
typedef __attribute__((ext_vector_type(8))) float v8f;

// ---------------------------------------------------------------- zero
__global__ void gcn_zero_f32(float* __restrict__ p, long long n) {
    long long i = (long long)blockIdx.x * blockDim.x + threadIdx.x;
    if (i < n) p[i] = 0.0f;
}

// ---------------------------------------------------------------- degrees
__global__ void gcn_degree(const int* __restrict__ src, const int* __restrict__ dst,
                           float* __restrict__ degO, float* __restrict__ degI, int E) {
    int e = blockIdx.x * blockDim.x + threadIdx.x;
    if (e < E) {
        atomicAdd(&degO[src[e]], 1.0f);
        atomicAdd(&degI[dst[e]], 1.0f);
    }
}

// ---------------------------------------------------------------- deg -> clamp(deg,1)^-0.5 (in place)
__global__ void gcn_inv_sqrt(float* __restrict__ d, long long n) {
    long long i = (long long)blockIdx.x * blockDim.x + threadIdx.x;
    if (i < n) {
        float v = d[i];
        v = v < 1.0f ? 1.0f : v;
        d[i] = 1.0f / sqrtf(v);
    }
}

// ---------------------------------------------------------------- edge scatter
// One wave32 per edge. Lane L handles channels [2L, 2L+1] of H=64.
// msg = h[src] * inv_sqrt_out[src];  agg[dst] += msg  (f32 atomics, L2-resident:
// node matrix is 25.6 MB << 192 MB L2, so gathers and atomics stay on-chip)
__global__ void gcn_edge_scatter(const float* __restrict__ hin,
                                 const float* __restrict__ invO,
                                 const int* __restrict__ src,
                                 const int* __restrict__ dst,
                                 float* __restrict__ agg, int E) {
    int wave = (int)((blockIdx.x * (long long)blockDim.x + threadIdx.x) >> 5);
    int lane = threadIdx.x & (WAVE - 1);
    if (wave >= E) return;                 // wave-uniform exit
    int s = src[wave];
    int d = dst[wave];
    float sc = invO[s];
    const float2* hp = (const float2*)(hin + (long long)s * Hh);
    float2 v = hp[lane];
    float* ap = agg + (long long)d * Hh + lane * 2;
    atomicAdd(ap,     v.x * sc);
    atomicAdd(ap + 1, v.y * sc);
}

// ---------------------------------------------------------------- WMMA GEMM + bias + ReLU
// hout[n,:] = relu( (agg[n,:] * inv_sqrt_in[n]) @ W + b )
// One wave computes a 16-row x 64-col panel = 4 accumulator tiles of 16x16 f32.
// K-loop: 16 steps of V_WMMA_F32_16X16X4_F32.
//
// f32 A (16x4) layout: lane L holds row M=L%16; VGPR0/1 hold K = (L>>4)*2 + {0,1}.
// f32 B (4x16) layout: lane L holds col N=L%16; VGPR0/1 hold K = (L>>4)*2 + {0,1}.
// f32 C/D (16x16)    : VGPR r, lanes 0-15 -> M=r, lanes 16-31 -> M=r+8; N = L%16.
//
// W is staged in LDS in fragment-ready interleaved layout:
//   Ws[(k>>1)*128 + n*2 + (k&1)] = W[k][n]
// so each B fragment is ONE ds_load_b64 into an even-aligned VGPR pair
// (no v_mov shuffles before the WMMA).
__global__ void gcn_gemm_relu(const float* __restrict__ agg,
                              const float* __restrict__ invI,
                              const float* __restrict__ W,
                              const float* __restrict__ bias,
                              float* __restrict__ hout,
                              int nPanels, int nNodes) {
    __shared__ float Ws[Hh * Hh];          // 16 KB of 320 KB LDS, swizzled
    for (int i = threadIdx.x; i < Hh * Hh; i += blockDim.x) {
        int k = i >> 6;                    // i / 64
        int n = i & 63;                    // i % 64
        Ws[(k >> 1) * 128 + n * 2 + (k & 1)] = W[i];
    }
    __syncthreads();

    int wavesPerBlock = blockDim.x >> 5;
    int wave = blockIdx.x * wavesPerBlock + (threadIdx.x >> 5);
    int lane = threadIdx.x & (WAVE - 1);
    if (wave >= nPanels) return;           // wave-uniform: live waves keep EXEC all-ones

    int row0 = wave * 16;
    int m    = lane & 15;
    int half = lane >> 4;                  // selects K pair for A/B fragments

    int arow = row0 + m;
    if (arow > nNodes - 1) arow = nNodes - 1;   // safe clamp (N%16==0 here anyway)
    float scale = invI[arow];
    const float* Ap = agg + (long long)arow * Hh;

    const v2f* Wp = (const v2f*)Ws;        // pair-indexed: Wp[(k>>1)*64 + n]

    v8f acc0 = {}, acc1 = {}, acc2 = {}, acc3 = {};

    for (int k0 = 0; k0 < Hh; k0 += 4) {
        int kk = k0 + half * 2;            // even
        v2f a;
        a.x = Ap[kk]     * scale;
        a.y = Ap[kk + 1] * scale;

        int bbase = (kk >> 1) * 64 + m;    // v2f index of (K=kk..kk+1, N=m)
        v2f b0 = Wp[bbase +  0];
        v2f b1 = Wp[bbase + 16];
        v2f b2 = Wp[bbase + 32];
        v2f b3 = Wp[bbase + 48];

        acc0 = __builtin_amdgcn_wmma_f32_16x16x4_f32(false, a, false, b0, (short)0, acc0, false, false);
        acc1 = __builtin_amdgcn_wmma_f32_16x16x4_f32(false, a, false, b1, (short)0, acc1, false, false);
        acc2 = __builtin_amdgcn_wmma_f32_16x16x4_f32(false, a, false, b2, (short)0, acc2, false, false);
        acc3 = __builtin_amdgcn_wmma_f32_16x16x4_f32(false, a, false, b3, (short)0, acc3, false, false);
    }

    // epilogue: bias + ReLU, C/D layout stores
    v8f accs[4] = {acc0, acc1, acc2, acc3};
#pragma unroll
    for (int t = 0; t < 4; t++) {
        int n = t * 16 + m;
        float bv = bias[n];
#pragma unroll
        for (int r = 0; r < 8; r++) {
            int rr = row0 + r + half * 8;
            if (rr < nNodes) {
                float v = accs[t][r] + bv;
                hout[(long long)rr * Hh + n] = v > 0.0f ? v : 0.0f;
            }
        }
    }
}

// ---------------------------------------------------------------- launch
extern "C" void kernel_launch(void* const* d_in, const int* in_sizes, int n_in,
                              void* d_out, int out_size, void* d_ws, size_t ws_size,
                              hipStream_t stream) {
    const float* features = (const float*)d_in[0];
    const float* W        = (const float*)d_in[1];
    const float* bias     = (const float*)d_in[2];
    const int*   src      = (const int*)d_in[3];
    const int*   dst      = (const int*)d_in[4];

    const int N = in_sizes[0] / Hh;
    const int E = in_sizes[3];

    // workspace layout (floats)
    float* degO = (float*)d_ws;                 // [N]  -> becomes inv_sqrt_out
    float* degI = degO + N;                     // [N]  -> becomes inv_sqrt_in
    float* agg  = degI + N;                     // [N*H]
    float* h1   = agg + (long long)N * Hh;      // [N*H]
    float* out  = (float*)d_out;

    const int TB = 256;
    const long long twoN = 2LL * N;
    const long long NH   = (long long)N * Hh;

    // degrees + normalization (recomputed every call: deterministic, no stale state)
    gcn_zero_f32<<<(int)((twoN + TB - 1) / TB), TB, 0, stream>>>(degO, twoN);
    gcn_degree<<<(E + TB - 1) / TB, TB, 0, stream>>>(src, dst, degO, degI, E);
    gcn_inv_sqrt<<<(int)((twoN + TB - 1) / TB), TB, 0, stream>>>(degO, twoN);

    const int nPanels    = (N + 15) / 16;
    const int gemmBlocks = (nPanels + (TB / WAVE) - 1) / (TB / WAVE);
    const int edgeBlocks = (int)(((long long)E * WAVE + TB - 1) / TB);

    // layer 1: features -> h1
    gcn_zero_f32<<<(int)((NH + TB - 1) / TB), TB, 0, stream>>>(agg, NH);
    gcn_edge_scatter<<<edgeBlocks, TB, 0, stream>>>(features, degO, src, dst, agg, E);
    gcn_gemm_relu<<<gemmBlocks, TB, 0, stream>>>(agg, degI, W, bias, h1, nPanels, N);

    // layer 2: h1 -> out
    gcn_zero_f32<<<(int)((NH + TB - 1) / TB), TB, 0, stream>>>(agg, NH);
    gcn_edge_scatter<<<edgeBlocks, TB, 0, stream>>>(h1, degO, src, dst, agg, E);
    gcn_gemm_relu<<<gemmBlocks, TB, 0, stream>>>(agg, degI, W, bias, out, nPanels, N);
}